// FullyConnected_83047487635670
// MI455X (gfx1250) — compile-verified
//
#include <hip/hip_runtime.h>
#include <cstdint>
#include <math.h>

#define IN_F   512
#define OUT_F  512

#define BR   32   // rows of x per block
#define BO   64   // output neurons per block
#define BK   32   // K slice per pipeline stage
#define LSTR 36   // LDS row stride in floats: 32 + 4 pad (144B = 16B aligned, bank skew 4*9)

// One wave instruction: each active lane copies 16B global -> LDS at its per-lane
// LDS byte offset. Tracked by ASYNCcnt; loads complete in order (ISA 5.7.1 / ch.10).
__device__ __forceinline__ void async_cp16(uint32_t lds_byte_off, const float* gptr) {
    asm volatile("global_load_async_to_lds_b128 %0, %1, off"
                 :: "v"(lds_byte_off),
                    "v"((unsigned long long)(uintptr_t)gptr)
                 : "memory");
}
// Steady-state: 3 newer async ops (next slice) may remain in flight.
__device__ __forceinline__ void wait_async_le3() { asm volatile("s_wait_asynccnt 0x3" ::: "memory"); }
__device__ __forceinline__ void wait_async_le0() { asm volatile("s_wait_asynccnt 0x0" ::: "memory"); }

__global__ __launch_bounds__(256)
void mam_fc_cdna5_kernel(const float* __restrict__ x,
                         const float* __restrict__ W,
                         const float* __restrict__ b,
                         float* __restrict__ out) {
    __shared__ __align__(16) float Xs[2][BR][LSTR];   // 2 x 32 x 36 floats
    __shared__ __align__(16) float Ws[2][BO][LSTR];   // 2 x 64 x 36 floats  (27.6KB total)

    const int tid = threadIdx.x;
    const int tx  = tid & 15;   // out lane  -> outputs {tx, tx+16, tx+32, tx+48}
    const int ty  = tid >> 4;   // row lane  -> rows    {ty, ty+16}

    const int outBase = blockIdx.x * BO;
    const int rowBase = blockIdx.y * BR;

    // LDS byte offset of a generic __shared__ pointer == low 32 bits (ISA 10.2 aperture rule)
    const uint32_t xsB[2] = { (uint32_t)(uintptr_t)&Xs[0][0][0], (uint32_t)(uintptr_t)&Xs[1][0][0] };
    const uint32_t wsB[2] = { (uint32_t)(uintptr_t)&Ws[0][0][0], (uint32_t)(uintptr_t)&Ws[1][0][0] };

    // Issue one K-slice: exactly 3 async wave-instructions per wave (1 X chunk + 2 W chunks per thread).
    auto issue_slice = [&](int kk, int s) {
        {   // X tile: 32 rows x 8 chunks = 256 chunks, 1 per thread
            const int r  = tid >> 3;           // 0..31
            const int kq = (tid & 7) << 2;     // 0..28 floats
            async_cp16(xsB[s] + (uint32_t)(r * (LSTR * 4) + kq * 4),
                       x + (size_t)(rowBase + r) * IN_F + kk + kq);
        }
#pragma unroll
        for (int c = 0; c < 2; ++c) {          // W tile: 64 rows x 8 chunks = 512 chunks, 2 per thread
            const int idx = c * 256 + tid;
            const int r   = idx >> 3;          // 0..63
            const int kq  = (idx & 7) << 2;    // 0..28 floats
            async_cp16(wsB[s] + (uint32_t)(r * (LSTR * 4) + kq * 4),
                       W + (size_t)(outBase + r) * IN_F + kk + kq);
        }
    };

    float mx[2][4], mn[2][4];
#pragma unroll
    for (int i = 0; i < 2; ++i)
#pragma unroll
        for (int j = 0; j < 4; ++j) { mx[i][j] = -INFINITY; mn[i][j] = INFINITY; }

    // ---- software pipeline: prologue fill of buffer 0 ----
    issue_slice(0, 0);

    int buf = 0;
    for (int kk = 0; kk < IN_F; kk += BK) {
        if (kk + BK < IN_F) {
            issue_slice(kk + BK, buf ^ 1);   // next slice in flight behind compute
            wait_async_le3();                // retire only the oldest 3 (current slice)
        } else {
            wait_async_le0();                // drain
        }
        __syncthreads();                     // current slice visible to all waves

        const float (*Xb)[LSTR] = Xs[buf];
        const float (*Wb)[LSTR] = Ws[buf];

#pragma unroll 4
        for (int k0 = 0; k0 < BK; k0 += 4) {
            float4 xv[2], wv[4];
            xv[0] = *reinterpret_cast<const float4*>(&Xb[ty     ][k0]);
            xv[1] = *reinterpret_cast<const float4*>(&Xb[ty + 16][k0]);
#pragma unroll
            for (int j = 0; j < 4; ++j)
                wv[j] = *reinterpret_cast<const float4*>(&Wb[tx + 16 * j][k0]);
#pragma unroll
            for (int i = 0; i < 2; ++i) {
#pragma unroll
                for (int j = 0; j < 4; ++j) {
                    float p;
                    p = xv[i].x * wv[j].x; mx[i][j] = fmaxf(mx[i][j], p); mn[i][j] = fminf(mn[i][j], p);
                    p = xv[i].y * wv[j].y; mx[i][j] = fmaxf(mx[i][j], p); mn[i][j] = fminf(mn[i][j], p);
                    p = xv[i].z * wv[j].z; mx[i][j] = fmaxf(mx[i][j], p); mn[i][j] = fminf(mn[i][j], p);
                    p = xv[i].w * wv[j].w; mx[i][j] = fmaxf(mx[i][j], p); mn[i][j] = fminf(mn[i][j], p);
                }
            }
        }
        __syncthreads();                     // slice consumed; its buffer may be refilled next iter
        buf ^= 1;
    }

    // ---- epilogue: C = max + min + bias ----
#pragma unroll
    for (int j = 0; j < 4; ++j) {
        const int o    = outBase + tx + 16 * j;
        const float bj = b[o];
#pragma unroll
        for (int i = 0; i < 2; ++i) {
            const int r = rowBase + ty + 16 * i;
            out[(size_t)r * OUT_F + o] = mx[i][j] + mn[i][j] + bj;
        }
    }
}

extern "C" void kernel_launch(void* const* d_in, const int* in_sizes, int n_in,
                              void* d_out, int out_size, void* d_ws, size_t ws_size,
                              hipStream_t stream) {
    const float* x = (const float*)d_in[0];   // (2*512, 512) flattened rows
    const float* W = (const float*)d_in[1];   // (512, 512)
    const float* b = (const float*)d_in[2];   // (512,)
    float* out = (float*)d_out;               // (1024, 512)

    const int rows = in_sizes[0] / IN_F;      // 1024
    dim3 grid(OUT_F / BO, rows / BR);         // (8, 32) = 256 blocks
    mam_fc_cdna5_kernel<<<grid, 256, 0, stream>>>(x, W, b, out);
}